// InvertibleConv1x1GridAlign_17961553232467
// MI455X (gfx1250) — compile-verified
//
#include <hip/hip_runtime.h>
#include <math.h>

typedef float v2f __attribute__((ext_vector_type(2)));
typedef float v8f __attribute__((ext_vector_type(8)));

#define BATCH 16
#define CHN   64
#define HH    160
#define WW    160
#define NBLK  161            // computation blocks per spatial dim
#define PLANE (HH * WW)      // 25600

__device__ __forceinline__ int clampi(int v, int lo, int hi) {
  return v < lo ? lo : (v > hi ? hi : v);
}

// ---------------- interior fast path ----------------
__device__ __forceinline__ void conv_tile_fast(const float* __restrict__ xb,
                                               float* __restrict__ ob,
                                               const float* __restrict__ Wp,
                                               int i, int j0, int lane) {
  const int nn  = lane & 15;   // N index (and A/M row) within 16
  const int khi = lane >> 4;   // selects K pair (+2) and M half (+8)

  v8f acc[4][2];
  const v8f vzero = {0.f, 0.f, 0.f, 0.f, 0.f, 0.f, 0.f, 0.f};
#pragma unroll
  for (int mt = 0; mt < 4; ++mt)
#pragma unroll
    for (int nt = 0; nt < 2; ++nt)
      acc[mt][nt] = vzero;

  const int abase = khi * 128 + nn * 2;
  const int sbase = (i - 1) * WW + (j0 - 1) + nn;
  const int vload = sbase + (khi ? (WW - 2 * PLANE) : 0);

#pragma unroll
  for (int t = 0; t < 16; ++t) {
    v2f bf[2];
#pragma unroll
    for (int nt = 0; nt < 2; ++nt) {
      bf[nt].x = xb[vload + nt * 16 + (4 * t + 3) * PLANE];
      bf[nt].y = xb[vload + nt * 16 + (4 * t + 2) * PLANE + 1];
    }
#pragma unroll
    for (int mt = 0; mt < 4; ++mt) {
      const v2f af = *(const v2f*)&Wp[abase + t * 256 + mt * 32];
      acc[mt][0] = __builtin_amdgcn_wmma_f32_16x16x4_f32(
          false, af, false, bf[0], (short)0, acc[mt][0], false, false);
      acc[mt][1] = __builtin_amdgcn_wmma_f32_16x16x4_f32(
          false, af, false, bf[1], (short)0, acc[mt][1], false, false);
    }
  }

  const int vstore = sbase + khi * 8 * PLANE;  // (m+8)^3 == (m^3)+8
#pragma unroll
  for (int mt = 0; mt < 4; ++mt)
#pragma unroll
    for (int nt = 0; nt < 2; ++nt)
#pragma unroll
      for (int r = 0; r < 8; ++r) {
        const int mlo = mt * 16 + r;
        const int fy = (r >> 1) & 1, fx = r & 1;
        ob[vstore + nt * 16 + (mlo ^ 3) * PLANE + fy * WW + fx] = acc[mt][nt][r];
      }
}

// ---------------- boundary path: hoisted masks, branchless loads ----------------
__device__ __forceinline__ void conv_tile_chk(const float* __restrict__ xb,
                                              float* __restrict__ ob,
                                              const float* __restrict__ Wp,
                                              int i, int j0, int lane) {
  const int nn  = lane & 15;
  const int khi = lane >> 4;

  // column validity / clamped columns (4 distinct positions per lane)
  bool cmask[2][2];
  int  colc[2][2];
#pragma unroll
  for (int nt = 0; nt < 2; ++nt)
#pragma unroll
    for (int gx = 0; gx < 2; ++gx) {
      const int col = j0 + nt * 16 + nn - 1 + gx;
      cmask[nt][gx] = (unsigned)col < (unsigned)WW;
      colc[nt][gx]  = clampi(col, 0, WW - 1);
    }
  // load row: single per-lane value i-1+khi
  const int  rowl   = i - 1 + khi;
  const bool rokl   = (unsigned)rowl < (unsigned)HH;
  const int  rowlc  = clampi(rowl, 0, HH - 1);
  // per-lane load voffsets (clamped => always-valid addresses)
  int  vb[2][2];
  bool okl[2][2];
#pragma unroll
  for (int nt = 0; nt < 2; ++nt)
#pragma unroll
    for (int gx = 0; gx < 2; ++gx) {
      vb[nt][gx]  = rowlc * WW + colc[nt][gx] + (khi ? -2 * PLANE : 0);
      okl[nt][gx] = rokl && cmask[nt][gx];
    }

  v8f acc[4][2];
  const v8f vzero = {0.f, 0.f, 0.f, 0.f, 0.f, 0.f, 0.f, 0.f};
#pragma unroll
  for (int mt = 0; mt < 4; ++mt)
#pragma unroll
    for (int nt = 0; nt < 2; ++nt)
      acc[mt][nt] = vzero;

  const int abase = khi * 128 + nn * 2;

#pragma unroll
  for (int t = 0; t < 16; ++t) {
    v2f bf[2];
#pragma unroll
    for (int nt = 0; nt < 2; ++nt) {
      const float vx = xb[vb[nt][0] + (4 * t + 3) * PLANE];  // unconditional, clamped
      const float vy = xb[vb[nt][1] + (4 * t + 2) * PLANE];
      bf[nt].x = okl[nt][0] ? vx : 0.0f;
      bf[nt].y = okl[nt][1] ? vy : 0.0f;
    }
#pragma unroll
    for (int mt = 0; mt < 4; ++mt) {
      const v2f af = *(const v2f*)&Wp[abase + t * 256 + mt * 32];
      acc[mt][0] = __builtin_amdgcn_wmma_f32_16x16x4_f32(
          false, af, false, bf[0], (short)0, acc[mt][0], false, false);
      acc[mt][1] = __builtin_amdgcn_wmma_f32_16x16x4_f32(
          false, af, false, bf[1], (short)0, acc[mt][1], false, false);
    }
  }

  // store masks: 2 row bools x (nt,fx) column bools, all hoisted
  bool rs[2];
  rs[0] = (unsigned)(i - 1) < (unsigned)HH;
  rs[1] = (unsigned)i < (unsigned)HH;
  const int vstore = (i - 1) * WW + (j0 - 1) + nn + khi * 8 * PLANE;
#pragma unroll
  for (int mt = 0; mt < 4; ++mt)
#pragma unroll
    for (int nt = 0; nt < 2; ++nt)
#pragma unroll
      for (int r = 0; r < 8; ++r) {
        const int mlo = mt * 16 + r;
        const int fy = (r >> 1) & 1, fx = r & 1;
        if (rs[fy] && cmask[nt][fx])
          ob[vstore + nt * 16 + (mlo ^ 3) * PLANE + fy * WW + fx] = acc[mt][nt][r];
      }
}

// Block = 192 threads = 6 waves; wave w handles j-tile jt=w.
// Grid = (161, 16): blockIdx.x = i, blockIdx.y = batch. No integer division.
__global__ __launch_bounds__(192)
void conv_wmma_kernel(const float* __restrict__ x,
                      const float* __restrict__ weight,
                      float* __restrict__ out) {
  // Packed A layout: Wp[(k>>1)*128 + m*2 + (k&1)] = W[m][k]
  __shared__ float Wp[CHN * CHN];
  const int tid = threadIdx.x;
  if (tid < 128) {
#pragma unroll
    for (int e = 0; e < 32; ++e) {
      const int idx = e * 128 + tid;   // 4096 elements
      const int m = idx >> 6, k = idx & 63;
      Wp[(k >> 1) * 128 + m * 2 + (k & 1)] = weight[idx];
    }
  }
  __syncthreads();

  const int lane = tid & 31;
  const int jt   = tid >> 5;        // wave id == j-tile id (0..5)
  const int i    = blockIdx.x;      // 0..160
  const int b    = blockIdx.y;      // 0..15
  const int j0   = jt * 32;

  const float* xb = x   + (size_t)b * CHN * PLANE;
  float*       ob = out + (size_t)b * CHN * PLANE;

  // Interior iff rows i-1..i and cols j0-1..j0+31 are all in-bounds.
  const bool interior = (i >= 1) && (i <= HH - 1) && (jt >= 1) && (jt <= 4);
  if (interior)
    conv_tile_fast(xb, ob, Wp, i, j0, lane);
  else
    conv_tile_chk(xb, ob, Wp, i, j0, lane);
}

// ---- dlogdet: 9 submatrix slogdets via LU with partial pivoting ----
__global__ __launch_bounds__(64)
void logdet_kernel(const float* __restrict__ weight,
                   const float* __restrict__ logdet_in,
                   float* __restrict__ logdet_out) {
  __shared__ float A[64][64];
  __shared__ int   idxs[64];
  __shared__ int   s_piv;
  __shared__ float s_total;
  const int tid = threadIdx.x;
  const unsigned char patterns[9] = {0xF, 0xC, 0x3, 0xA, 0x5, 0x8, 0x4, 0x2, 0x1};
  const int pixels[9] = {25281, 159, 159, 159, 159, 1, 1, 1, 1};
  if (tid == 0) s_total = 0.0f;

  for (int mi = 0; mi < 9; ++mi) {
    const int pat = patterns[mi];
    const int n = 16 * __popc(pat);
    if (tid == 0) {
      int cnt = 0;
      for (int c = 0; c < 64; ++c)
        if ((pat >> (c & 3)) & 1) idxs[cnt++] = c;
    }
    __syncthreads();
    for (int e = tid; e < n * n; e += 64) {
      int r = e / n, c = e % n;
      A[r][c] = weight[idxs[r] * 64 + idxs[c]];
    }
    __syncthreads();
    float local = 0.0f;
    for (int k = 0; k < n; ++k) {
      if (tid == 0) {
        int p = k; float best = fabsf(A[k][k]);
        for (int r = k + 1; r < n; ++r) {
          float v = fabsf(A[r][k]);
          if (v > best) { best = v; p = r; }
        }
        s_piv = p;
      }
      __syncthreads();
      const int p = s_piv;
      if (p != k && tid < n) {
        float tmp = A[k][tid]; A[k][tid] = A[p][tid]; A[p][tid] = tmp;
      }
      __syncthreads();
      if (tid == 0) local += logf(fabsf(A[k][k]));
      if (tid > k && tid < n) {
        float f = A[tid][k] / A[k][k];
        for (int c = k; c < n; ++c) A[tid][c] -= f * A[k][c];
      }
      __syncthreads();
    }
    if (tid == 0) s_total += local * (float)pixels[mi];
    __syncthreads();
  }
  if (tid < 16) logdet_out[tid] = logdet_in[tid] + s_total;
}

extern "C" void kernel_launch(void* const* d_in, const int* in_sizes, int n_in,
                              void* d_out, int out_size, void* d_ws, size_t ws_size,
                              hipStream_t stream) {
  const float* x      = (const float*)d_in[0];   // (16,64,160,160) f32
  const float* logdet = (const float*)d_in[1];   // (16,) f32
  const float* weight = (const float*)d_in[2];   // (64,64) f32
  float* out        = (float*)d_out;
  float* out_logdet = out + (size_t)BATCH * CHN * PLANE;

  // grid: x = 161 block-rows, y = 16 batches; 6 waves/block = 6 j-tiles
  conv_wmma_kernel<<<dim3(NBLK, BATCH), dim3(192), 0, stream>>>(x, weight, out);
  logdet_kernel<<<dim3(1), dim3(64), 0, stream>>>(weight, logdet, out_logdet);
}